// PairTabModel_86629490361084
// MI455X (gfx1250) — compile-verified
//
#include <hip/hip_runtime.h>
#include <hip/hip_bf16.h>

// ---------------- problem constants (from reference) ----------------
#define NF      2
#define NLOC    3072
#define NALL    4096
#define NNEI    128
#define NTYPES  4
#define NSPLINE 1024
#define RCUT    6.0f

#define ATOMS_PER_BLOCK 128
#define CHUNKS_PER_FRAME (NLOC / ATOMS_PER_BLOCK)   // 24
#define BLOCK_THREADS 1024                          // 32 wave32 waves

// LDS layout: coord[f] (NALL*3 floats = 48KB) then atype[f] (NALL ints = 16KB)
#define LDS_COORD_BYTES (NALL * 3 * 4)
#define LDS_ATYPE_BYTES (NALL * 4)
#define LDS_TOTAL_BYTES (LDS_COORD_BYTES + LDS_ATYPE_BYTES)  // 65536

typedef unsigned int v4u __attribute__((ext_vector_type(4)));
typedef int          v8i __attribute__((ext_vector_type(8)));
typedef int          v4i __attribute__((ext_vector_type(4)));

// ---------------------------------------------------------------------
// Issue one TDM 1-D copy of `ndwords` DWORDs from global -> LDS.
// Descriptor packing per CDNA5 ISA §8 (D# groups):
//   group0: count=1 | lds_addr | global_addr[56:0] | type=2
//   group1: data_size=4B; tensor_dim0=ndwords; tile_dim0=ndwords (1-D tile,
//           tile_dim1=0 => unused); tensor_dim0_stride=ndwords
//   groups 2/3: zero (tile_dim3/4 = 0 => unused)
// ---------------------------------------------------------------------
__device__ __forceinline__ void tdm_load_1d(unsigned lds_byte_off,
                                            const void* gptr,
                                            unsigned ndwords) {
  unsigned long long ga = (unsigned long long)gptr;

  v4u g0;
  g0[0] = 1u;                                        // count=1 (valid, user)
  g0[1] = lds_byte_off;                              // lds_addr (bytes)
  g0[2] = (unsigned)(ga & 0xffffffffu);              // global_addr[31:0]
  g0[3] = (unsigned)((ga >> 32) & 0x01ffffffu)       // global_addr[56:32]
        | (2u << 30);                                // type = 2 ("image")

  v8i g1;
  g1[0] = (int)(2u << 16);                           // wg_mask=0, data_size=4B
  g1[1] = (int)((ndwords & 0xffffu) << 16);          // tensor_dim0[15:0]
  g1[2] = (int)(((ndwords >> 16) & 0xffffu)          // tensor_dim0[31:16]
        | (1u << 16));                               // tensor_dim1 = 1
  g1[3] = (int)((ndwords & 0xffffu) << 16);          // tile_dim0 (<=65535)
  g1[4] = 0;                                         // tile_dim1=0, tile_dim2=0
  g1[5] = (int)ndwords;                              // tensor_dim0_stride[31:0]
  g1[6] = 0;
  g1[7] = 0;

  v4i z4 = {0, 0, 0, 0};
#if __has_include(<hip/amd_detail/amd_gfx1250_TDM.h>)
  v8i z8 = {0, 0, 0, 0, 0, 0, 0, 0};
  __builtin_amdgcn_tensor_load_to_lds(g0, g1, z4, z4, z8, 0);   // clang-23 form
#else
  __builtin_amdgcn_tensor_load_to_lds(g0, g1, z4, z4, 0);       // ROCm 7.2 form
#endif
}

__global__ __launch_bounds__(BLOCK_THREADS)
void PairTab_gfx1250_kernel(const float* __restrict__ coord,     // [NF,NALL,3]
                            const int*   __restrict__ atype,     // [NF,NALL]
                            const int*   __restrict__ nlist,     // [NF,NLOC,NNEI]
                            const float* __restrict__ tab_info,  // [3]
                            const float* __restrict__ tab_data,  // [NT,NT,NSPLINE,4]
                            float*       __restrict__ out)       // [NF,NLOC]
{
  extern __shared__ char smem[];
  float* s_coord = (float*)smem;
  int*   s_atype = (int*)(smem + LDS_COORD_BYTES);

  const int f     = blockIdx.x / CHUNKS_PER_FRAME;
  const int chunk = blockIdx.x % CHUNKS_PER_FRAME;

  // ---- wave 0: DMA-stage this frame's coords + types into LDS via TDM ----
  if ((threadIdx.x >> 5) == 0) {
    unsigned lds_base = (unsigned)(unsigned long long)(void*)smem; // LDS offset (low 32 bits of flat addr)
    tdm_load_1d(lds_base, coord + (size_t)f * NALL * 3, NALL * 3);
    tdm_load_1d(lds_base + LDS_COORD_BYTES, atype + (size_t)f * NALL, NALL);
    __builtin_amdgcn_s_wait_tensorcnt((short)0);
  }

  // ---- warm L2 with the 256KB spline table: 1024 thr x 256B ----
  {
    const char* tb = (const char*)tab_data;
    __builtin_prefetch(tb + (size_t)threadIdx.x * 256, 0, 1);
    __builtin_prefetch(tb + (size_t)threadIdx.x * 256 + 128, 0, 1);
  }

  __syncthreads();   // publish TDM-written LDS to all 32 waves

  const float rmin   = tab_info[0];
  const float hh     = tab_info[1];
  const float inv_hh = 1.0f / hh;                    // one divide, hoisted
  const float upper  = rmin + (float)NSPLINE * hh;   // spline upper bound
  const float4* tabv = (const float4*)tab_data;      // [NT*NT*NSPLINE] float4

  const int wave = threadIdx.x >> 5;
  const int lane = threadIdx.x & 31;

  // 32 waves x 4 atoms each = 128 atoms per block; 32 lanes x 4 neighbors.
  for (int a = wave; a < ATOMS_PER_BLOCK; a += 32) {
    const int loc = chunk * ATOMS_PER_BLOCK + a;

    const float cx = s_coord[loc * 3 + 0];
    const float cy = s_coord[loc * 3 + 1];
    const float cz = s_coord[loc * 3 + 2];
    const int   it = s_atype[loc];

    // coalesced: lane L reads nlist[..., 4L..4L+3] as one b128
    const int4 nl =
        ((const int4*)(nlist + ((size_t)f * NLOC + loc) * NNEI))[lane];
    const int js[4] = {nl.x, nl.y, nl.z, nl.w};

    float acc = 0.0f;
#pragma unroll
    for (int k = 0; k < 4; ++k) {
      const int  j     = js[k];
      const bool valid = (j != -1);
      const int  mj    = valid ? j : 0;

      const float dx = cx - s_coord[mj * 3 + 0];
      const float dy = cy - s_coord[mj * 3 + 1];
      const float dz = cz - s_coord[mj * 3 + 2];
      const float rr = sqrtf(fmaxf(dx * dx + dy * dy + dz * dz, 1e-30f));
      const int   jt = s_atype[mj];

      const float uu = valid ? (rr - rmin) * inv_hh : (float)(NSPLINE + 1);
      const int idx = (int)uu;                 // trunc toward zero (== astype)
      const float t = uu - (float)idx;
      int cidx = idx;
      cidx = cidx < 0 ? 0 : cidx;
      cidx = cidx > (NSPLINE - 1) ? (NSPLINE - 1) : cidx;

      const float4 c =
          tabv[((it * NTYPES + jt) * NSPLINE) + cidx];     // 16B L2 gather
      const float e = ((c.x * t + c.y) * t + c.z) * t + c.w;

      const bool beyond = (rr >= RCUT) || (rr >= upper);
      const bool zero   = (idx > NSPLINE) || beyond || (!valid);
      acc += zero ? 0.0f : e;
    }

    // wave32 lane reduction
#pragma unroll
    for (int off = 16; off > 0; off >>= 1)
      acc += __shfl_xor(acc, off, 32);

    if (lane == 0)
      out[(size_t)f * NLOC + loc] = 0.5f * acc;
  }
}

extern "C" void kernel_launch(void* const* d_in, const int* in_sizes, int n_in,
                              void* d_out, int out_size, void* d_ws, size_t ws_size,
                              hipStream_t stream) {
  const float* coord    = (const float*)d_in[0];
  const int*   atype    = (const int*)d_in[1];
  const int*   nlist    = (const int*)d_in[2];
  const float* tab_info = (const float*)d_in[3];
  const float* tab_data = (const float*)d_in[4];
  float*       out      = (float*)d_out;

  dim3 grid(NF * CHUNKS_PER_FRAME);     // 48 blocks
  dim3 block(BLOCK_THREADS);            // 32 waves of 32
  PairTab_gfx1250_kernel<<<grid, block, LDS_TOTAL_BYTES, stream>>>(
      coord, atype, nlist, tab_info, tab_data, out);
}